// MultiLayerGCN_83184926589196
// MI455X (gfx1250) — compile-verified
//
#include <hip/hip_runtime.h>
#include <cstdint>
#include <cstddef>

#define DF 128  // feature dimension (fixed by the reference)

typedef __attribute__((ext_vector_type(2))) float v2f;
typedef __attribute__((ext_vector_type(8))) float v8f;

__device__ __forceinline__ void atomic_add_f32(float* p, float v) {
  (void)__hip_atomic_fetch_add(p, v, __ATOMIC_RELAXED, __HIP_MEMORY_SCOPE_AGENT);
}

// ---------------- degree / symmetric norm ----------------
__global__ void k_init_deg(float* __restrict__ deg, int n) {
  int i = blockIdx.x * blockDim.x + threadIdx.x;
  if (i < n) deg[i] = 1.0f;  // self-loop contribution
}

__global__ void k_deg_edges(const int* __restrict__ dst, float* __restrict__ deg, int nE) {
  int e = blockIdx.x * blockDim.x + threadIdx.x;
  if (e < nE) atomic_add_f32(&deg[dst[e]], 1.0f);
}

__global__ void k_rsqrt(float* __restrict__ d, int n) {
  int i = blockIdx.x * blockDim.x + threadIdx.x;
  if (i < n) d[i] = rsqrtf(d[i]);  // deg >= 1 always (self-loops)
}

// ---------------- H = X * W via V_WMMA_F32_16X16X4_F32 ----------------
// Each wave owns a 16-row x 128-col output slab: A fragment loaded once per
// K-step and reused by 8 WMMAs (8 x v8f accumulators = 64 VGPRs, fine in
// wave32). Slab guard is wave-uniform -> EXEC all-ones for every WMMA.
__global__ void k_gemm_wmma(const float* __restrict__ X, const float* __restrict__ W,
                            float* __restrict__ H, int nSlabs) {
  const int lane = threadIdx.x & 31;
  const int slab = blockIdx.x * (blockDim.x >> 5) + (threadIdx.x >> 5);
  if (slab >= nSlabs) return;                 // uniform per wave
  const int row0 = slab << 4;
  const int m    = lane & 15;                 // M index (A) / N index (B,C,D)
  const int koff = (lane >> 4) << 1;          // lanes 0-15 -> K+0/1, 16-31 -> K+2/3

  const float* xrow = X + (size_t)(row0 + m) * DF;
  v8f c[8] = {};
  for (int k = 0; k < DF; k += 4) {
    v2f a;
    a.x = xrow[k + koff];
    a.y = xrow[k + koff + 1];
    const float* wr0 = W + (size_t)(k + koff) * DF + m;
    const float* wr1 = wr0 + DF;
#pragma unroll
    for (int t = 0; t < 8; ++t) {             // 8 column tiles, A reused
      v2f b;
      b.x = wr0[t * 16];
      b.y = wr1[t * 16];
      c[t] = __builtin_amdgcn_wmma_f32_16x16x4_f32(false, a, false, b, (short)0,
                                                   c[t], false, false);
    }
  }
  // C/D layout: lanes 0-15 hold rows M=0..7 in v0..v7, lanes 16-31 hold M=8..15
  const int rbase = row0 + ((lane >> 4) << 3);
#pragma unroll
  for (int t = 0; t < 8; ++t)
#pragma unroll
    for (int r = 0; r < 8; ++r)
      H[(size_t)(rbase + r) * DF + t * 16 + m] = c[t][r];
}

// ---------------- aggregation ----------------
// AGG initialized with the self-loop term: agg = h * dis[node]^2
// (full overwrite -> no zero-fill pass, no atomics for the 100k self-loops)
__global__ void k_selfloop(const float4* __restrict__ H4, const float* __restrict__ dis,
                           float* __restrict__ agg, size_t total4) {
  size_t i = (size_t)blockIdx.x * blockDim.x + threadIdx.x;
  if (i < total4) {
    float dn = dis[i >> 5];                    // 32 float4 per node row
    float s  = dn * dn;
    float4 v = H4[i];
    float4* o = (float4*)agg + i;
    *o = make_float4(v.x * s, v.y * s, v.z * s, v.w * s);
  }
}

// One wave per edge: 32 lanes x float4 = one coalesced 512B feature row.
// H (51.2 MB) and AGG (51.2 MB) both sit in the 192 MB L2: gathers mostly hit
// L2 and the fp32 atomics resolve at the L2 atomic units.
// Edge id is forced into an SGPR so src/dst/dis loads go down the scalar
// (SMEM/KMcnt) path, keeping the vector memory path free for payload+atomics.
__global__ void k_edge_scatter(const float4* __restrict__ H4, const int* __restrict__ src,
                               const int* __restrict__ dst, const float* __restrict__ dis,
                               float* __restrict__ agg, int nE) {
  const int lane = threadIdx.x & 31;
  int e = __builtin_amdgcn_readfirstlane(
      (int)(blockIdx.x * (blockDim.x >> 5) + (threadIdx.x >> 5)));
  if (e >= nE) return;                         // uniform per wave
  int s = src[e], d = dst[e];                  // scalar loads
  float nrm = dis[s] * dis[d];                 // scalar loads + scalar math
  float4 v = H4[(size_t)s * (DF / 4) + lane];
  float* o = agg + (size_t)d * DF + lane * 4;
  atomic_add_f32(o + 0, v.x * nrm);
  atomic_add_f32(o + 1, v.y * nrm);
  atomic_add_f32(o + 2, v.z * nrm);
  atomic_add_f32(o + 3, v.w * nrm);
}

__global__ void k_bias_relu(const float4* __restrict__ agg4, const float4* __restrict__ b4,
                            float4* __restrict__ out4, size_t total4) {
  size_t i = (size_t)blockIdx.x * blockDim.x + threadIdx.x;
  if (i < total4) {
    float4 v = agg4[i];
    float4 b = b4[i & (DF / 4 - 1)];
    out4[i] = make_float4(fmaxf(v.x + b.x, 0.0f), fmaxf(v.y + b.y, 0.0f),
                          fmaxf(v.z + b.z, 0.0f), fmaxf(v.w + b.w, 0.0f));
  }
}

// ---------------- host-side orchestration ----------------
extern "C" void kernel_launch(void* const* d_in, const int* in_sizes, int n_in,
                              void* d_out, int out_size, void* d_ws, size_t ws_size,
                              hipStream_t stream) {
  const float* x  = (const float*)d_in[0];
  const int*   ei = (const int*)d_in[1];
  const float* W1 = (const float*)d_in[2];
  const float* b1 = (const float*)d_in[3];
  const float* W2 = (const float*)d_in[4];
  const float* b2 = (const float*)d_in[5];
  float* out = (float*)d_out;

  const int N = in_sizes[0] / DF;   // 100000
  const int E = in_sizes[1] / 2;    // 1600000
  const int* srcP = ei;             // edge_index[0]
  const int* dstP = ei + E;         // edge_index[1]

  // workspace layout: dis | H | AGG
  char* ws = (char*)d_ws;
  size_t off = ((size_t)N * sizeof(float) + 255) & ~(size_t)255;
  float* dis = (float*)ws;
  float* H   = (float*)(ws + off);
  float* AGG = (float*)(ws + off + (size_t)N * DF * sizeof(float));

  const size_t total4 = (size_t)N * DF / 4;
  const int TB = 256;
  const int WPB = TB / 32;                                // waves per block
  const int nSlabs = N / 16;                              // 6250
  const int gN  = (N + TB - 1) / TB;
  const int gE  = (E + TB - 1) / TB;
  const int gT4 = (int)((total4 + TB - 1) / TB);
  const int gEW = (E + WPB - 1) / WPB;                    // wave per edge
  const int gG  = (nSlabs + WPB - 1) / WPB;               // wave per 16-row slab

  // normalization: deg (with self-loop) -> dis = deg^{-1/2}
  k_init_deg<<<gN, TB, 0, stream>>>(dis, N);
  k_deg_edges<<<gE, TB, 0, stream>>>(dstP, dis, E);
  k_rsqrt<<<gN, TB, 0, stream>>>(dis, N);

  // ---- layer 1: relu(D^-1/2 (A+I) D^-1/2 * (x W1) + b1) -> out ----
  k_gemm_wmma<<<gG, TB, 0, stream>>>(x, W1, H, nSlabs);
  k_selfloop<<<gT4, TB, 0, stream>>>((const float4*)H, dis, AGG, total4);
  k_edge_scatter<<<gEW, TB, 0, stream>>>((const float4*)H, srcP, dstP, dis, AGG, E);
  k_bias_relu<<<gT4, TB, 0, stream>>>((const float4*)AGG, (const float4*)b1,
                                      (float4*)out, total4);

  // ---- layer 2: same, input = out, result overwrites out ----
  k_gemm_wmma<<<gG, TB, 0, stream>>>(out, W2, H, nSlabs);
  k_selfloop<<<gT4, TB, 0, stream>>>((const float4*)H, dis, AGG, total4);
  k_edge_scatter<<<gEW, TB, 0, stream>>>((const float4*)H, srcP, dstP, dis, AGG, E);
  k_bias_relu<<<gT4, TB, 0, stream>>>((const float4*)AGG, (const float4*)b2,
                                      (float4*)out, total4);
}